// QuantumGraphConv_12043088298392
// MI455X (gfx1250) — compile-verified
//
#include <hip/hip_runtime.h>
#include <math.h>

typedef __attribute__((ext_vector_type(2))) float v2f;
typedef __attribute__((ext_vector_type(8))) float v8f;
typedef __attribute__((ext_vector_type(4))) int   v4i;

typedef __attribute__((address_space(1))) v4i gv4i;   // global int4
typedef __attribute__((address_space(3))) v4i lv4i;   // LDS int4

#define NQ      8
#define NLAYERS 4
#define BB      64
#define NN      512
#define HH      512
#define LDSS    516   // padded LDS row stride (516 % 64 == 4 -> spreads banks)
#define CHUNK   (16 * LDSS)

#ifndef __has_builtin
#define __has_builtin(x) 0
#endif
#if __has_builtin(__builtin_amdgcn_global_load_async_to_lds_b128) && \
    __has_builtin(__builtin_amdgcn_s_wait_asynccnt)
#define ASYNC_OK 1
#else
#define ASYNC_OK 0
#endif

#if ASYNC_OK
#define WAIT_ASYNC(n) __builtin_amdgcn_s_wait_asynccnt(n)
#else
#define WAIT_ASYNC(n)
#endif

// ---------------------------------------------------------------------------
// Stage one 16-row x 512-col f32 chunk (row stride grow_floats in global) into
// LDS (row stride LDSS) using async global->LDS B128 copies: 8 issues/thread,
// 256 threads, 16 threads per row, 256B stride per issue.
// ---------------------------------------------------------------------------
__device__ __forceinline__ void stage16(const float* __restrict__ gsrc,
                                        int grow_floats,
                                        float* __restrict__ lbuf, int tid)
{
    const int r  = tid >> 4;          // row 0..15
    const int cb = (tid & 15) << 4;   // byte column base 0..240
#if ASYNC_OK
    const char* g0 = (const char*)gsrc + (size_t)r * ((size_t)grow_floats * 4) + cb;
    char*       l0 = (char*)lbuf + r * (LDSS * 4) + cb;
    #pragma unroll
    for (int i = 0; i < 8; ++i) {
        __builtin_amdgcn_global_load_async_to_lds_b128(
            (gv4i*)(char*)(g0 + i * 256), (lv4i*)(l0 + i * 256), 0, 0);
    }
#else
    const float4* g0 = (const float4*)((const char*)gsrc +
                                       (size_t)r * ((size_t)grow_floats * 4) + cb);
    float4* l0 = (float4*)((char*)lbuf + r * (LDSS * 4) + cb);
    #pragma unroll
    for (int i = 0; i < 8; ++i) l0[i * 16] = g0[i * 16];
#endif
}

// ---------------------------------------------------------------------------
// Kernel 1: transpose W_gcn_w (HxH) into workspace: Wt[k][h] = W[h][k]
// ---------------------------------------------------------------------------
__global__ __launch_bounds__(256) void transpose512(const float* __restrict__ W,
                                                    float* __restrict__ Wt)
{
    __shared__ float tile[32][33];
    const int tx = threadIdx.x, ty = threadIdx.y;
    const int bx = blockIdx.x << 5, by = blockIdx.y << 5;
    #pragma unroll
    for (int r = ty; r < 32; r += 8)
        tile[r][tx] = W[(size_t)(by + r) * HH + bx + tx];
    __syncthreads();
    #pragma unroll
    for (int r = ty; r < 32; r += 8)
        Wt[(size_t)(bx + r) * HH + by + tx] = tile[tx][r];
}

// ---------------------------------------------------------------------------
// One GEMM stage: acc(16x64 per wave) += Alds(16x512, LDS) @ Bglob(512x512).
// B is streamed through a double-buffered LDS chunk (16 K-rows) with async
// copies one chunk ahead; WMMA f32 16x16x4 consumes A/B fragments from LDS.
// ---------------------------------------------------------------------------
__device__ __forceinline__ void gemm_stage(const float* __restrict__ Alds,
                                           const float* __restrict__ Bglob,
                                           float* __restrict__ sx,   // 2*CHUNK
                                           v8f acc[4],
                                           int tid, int l15, int hi2, int h0)
{
    int buf = 0;
    stage16(Bglob, HH, sx, tid);                       // chunk 0
    for (int jc = 0; jc < NN; jc += 16) {
        if (jc + 16 < NN) {
            stage16(Bglob + (size_t)(jc + 16) * HH, HH, sx + (buf ^ 1) * CHUNK, tid);
            WAIT_ASYNC(8);    // oldest 8 (current chunk) complete
        } else {
            WAIT_ASYNC(0);
        }
        __syncthreads();      // everyone's copies for current chunk visible

        const float* __restrict__ bb = sx + buf * CHUNK;
        #pragma unroll
        for (int kk = 0; kk < 16; kk += 4) {
            const v2f a = *(const v2f*)(Alds + l15 * LDSS + jc + kk + hi2);
            const float* __restrict__ br = bb + (kk + hi2) * LDSS + h0 + l15;
            #pragma unroll
            for (int t = 0; t < 4; ++t) {
                v2f bf;
                bf.x = br[t * 16];
                bf.y = br[LDSS + t * 16];
                acc[t] = __builtin_amdgcn_wmma_f32_16x16x4_f32(
                    false, a, false, bf, (short)0, acc[t], false, false);
            }
        }
        __syncthreads();      // done reading buf before it is refilled
        buf ^= 1;
    }
}

// ---------------------------------------------------------------------------
// Kernel 2: fused  out = gelu( (adj @ x[b]) @ W^T + b )  via f32 WMMA.
// Grid: (NN/16, BB). Block: 256 threads = 8 wave32; each wave owns a 16x64
// output strip (4 WMMA accumulators).
//
// f32 WMMA 16x16x4 operand mapping (ISA 7.12.2):
//   A (16x4): lanes 0-15 = M rows; per-lane v2f = A[M][kb], A[M][kb+1],
//             kb = 2*(lane>>4)
//   B (4x16): per-lane v2f = B[kb][n], B[kb+1][n], n = lane&15
//   C/D: VGPR v -> M = v + 8*(lane>>4), N = lane&15
// ---------------------------------------------------------------------------
__global__ __launch_bounds__(256) void gcn_fused_kernel(
    const float* __restrict__ x, const float* __restrict__ adj,
    const float* __restrict__ Wt, const float* __restrict__ bias,
    float* __restrict__ out)
{
    __shared__ __align__(16) float s_adj[16 * LDSS];  // A tile, stage 1
    __shared__ __align__(16) float s_S[16 * LDSS];    // stage-1 result = A tile, stage 2
    __shared__ __align__(16) float s_xb[2 * CHUNK];   // double-buffered B chunks

    const int b    = blockIdx.y;
    const int i0   = blockIdx.x << 4;
    const int tid  = threadIdx.x;
    const int wave = tid >> 5;
    const int lane = tid & 31;
    const int l15  = lane & 15;
    const int hi2  = (lane >> 4) << 1;
    const int h0   = wave << 6;

    // async-stage the 16x512 adjacency tile
    stage16(adj + (size_t)i0 * NN, NN, s_adj, tid);
    WAIT_ASYNC(0);
    __syncthreads();

    const float* __restrict__ xg = x + (size_t)b * NN * HH;

    const v8f zero8 = {0, 0, 0, 0, 0, 0, 0, 0};
    v8f acc[4] = {zero8, zero8, zero8, zero8};

    // ---- stage 1: S = adj_tile @ x[b] ----
    gemm_stage(s_adj, xg, s_xb, acc, tid, l15, hi2, h0);

    // spill accumulators to LDS (C/D layout -> row-major S)
    const int mbase = (lane >> 4) << 3;
    #pragma unroll
    for (int t = 0; t < 4; ++t) {
        const int col = h0 + t * 16 + l15;
        #pragma unroll
        for (int v = 0; v < 8; ++v)
            s_S[(mbase + v) * LDSS + col] = acc[t][v];
    }
    __syncthreads();

    #pragma unroll
    for (int t = 0; t < 4; ++t) acc[t] = zero8;

    // ---- stage 2: Y = S @ W^T  (Wt[k][h] = W[h][k]) ----
    gemm_stage(s_S, Wt, s_xb, acc, tid, l15, hi2, h0);

    // epilogue: bias + exact GELU, store
    #pragma unroll
    for (int t = 0; t < 4; ++t) {
        const int col = h0 + t * 16 + l15;
        const float bia = bias[col];
        #pragma unroll
        for (int v = 0; v < 8; ++v) {
            float val = acc[t][v] + bia;
            val = 0.5f * val * (1.0f + erff(val * 0.70710678118654752f));
            out[(size_t)b * NN * HH + (size_t)(i0 + mbase + v) * HH + col] = val;
        }
    }
}

// ---------------------------------------------------------------------------
// Kernel 3: quantum head. One block per batch; thread tid owns amplitude tid
// of the 2^8 statevector in LDS. Reads gelu'd TARGET row, overwrites it.
// ---------------------------------------------------------------------------
__global__ __launch_bounds__(256) void quantum_head_kernel(
    float* __restrict__ out,
    const float* __restrict__ pre_w, const float* __restrict__ pre_b,
    const float* __restrict__ post_w, const float* __restrict__ post_b,
    const float* __restrict__ qw)
{
    __shared__ float tgt[HH];
    __shared__ float sre[256], sim[256];
    __shared__ float ang[NQ], zs[NQ];

    const int b    = blockIdx.x;
    const int tid  = threadIdx.x;
    const int wave = tid >> 5;     // 8 waves == NQ
    const int lane = tid & 31;
    float* __restrict__ orow = out + (size_t)b * NN * HH;  // row TARGET == 0

    tgt[tid]       = orow[tid];
    tgt[tid + 256] = orow[tid + 256];
    __syncthreads();

    // x_qin[w] = tanh(tgt . pre_w[w] + pre_b[w]) * pi   (one wave per qubit)
    {
        float p = 0.0f;
        const float* pw = pre_w + wave * HH;
        for (int i = lane; i < HH; i += 32) p += tgt[i] * pw[i];
        #pragma unroll
        for (int off = 16; off > 0; off >>= 1) p += __shfl_xor(p, off, 32);
        if (lane == 0) ang[wave] = tanhf(p + pre_b[wave]) * 3.14159265358979f;
    }
    __syncthreads();

    // initial RY layer on |0..0> collapses to a real product state
    {
        float amp = 1.0f;
        #pragma unroll
        for (int w = 0; w < NQ; ++w) {
            const float half = 0.5f * ang[w];
            const int bit = (tid >> (NQ - 1 - w)) & 1;
            amp *= bit ? sinf(half) : cosf(half);
        }
        sre[tid] = amp;
        sim[tid] = 0.0f;
    }
    __syncthreads();

    for (int l = 0; l < NLAYERS; ++l) {
        // RX(weights[l,w]) on every qubit
        #pragma unroll
        for (int w = 0; w < NQ; ++w) {
            const int mask = 1 << (NQ - 1 - w);
            const float th = 0.5f * qw[l * NQ + w];
            const float c = cosf(th), s = sinf(th);
            if ((tid & mask) == 0) {
                const int p1 = tid | mask;
                const float a0r = sre[tid], a0i = sim[tid];
                const float a1r = sre[p1],  a1i = sim[p1];
                sre[tid] = c * a0r + s * a1i;   // c*a0 + (-i s)*a1
                sim[tid] = c * a0i - s * a1r;
                sre[p1]  = c * a1r + s * a0i;   // (-i s)*a0 + c*a1
                sim[p1]  = c * a1i - s * a0r;
            }
            __syncthreads();
        }
        // CNOT ring w -> (w+1) % NQ, applied sequentially
        #pragma unroll
        for (int w = 0; w < NQ; ++w) {
            const int mc = 1 << (NQ - 1 - w);
            const int mt = 1 << (NQ - 1 - ((w + 1) & (NQ - 1)));
            if ((tid & mc) != 0 && (tid & mt) == 0) {
                const int p1 = tid | mt;
                const float r0 = sre[tid], i0_ = sim[tid];
                sre[tid] = sre[p1]; sim[tid] = sim[p1];
                sre[p1]  = r0;      sim[p1]  = i0_;
            }
            __syncthreads();
        }
    }

    // probabilities (reuse tgt[0..255]) and <Z_w> = sum p * (1 - 2*bit_w)
    tgt[tid] = sre[tid] * sre[tid] + sim[tid] * sim[tid];
    __syncthreads();
    {
        float z = 0.0f;
        for (int idx = lane; idx < 256; idx += 32) {
            const int bit = (idx >> (NQ - 1 - wave)) & 1;
            z += bit ? -tgt[idx] : tgt[idx];
        }
        #pragma unroll
        for (int off = 16; off > 0; off >>= 1) z += __shfl_xor(z, off, 32);
        if (lane == 0) zs[wave] = z;
    }
    __syncthreads();

    // q_out = z @ post_w^T + post_b, overwrite TARGET row
    for (int h = tid; h < HH; h += 256) {
        float v = post_b[h];
        #pragma unroll
        for (int i = 0; i < NQ; ++i) v += zs[i] * post_w[h * NQ + i];
        orow[h] = v;
    }
}

// ---------------------------------------------------------------------------
extern "C" void kernel_launch(void* const* d_in, const int* in_sizes, int n_in,
                              void* d_out, int out_size, void* d_ws, size_t ws_size,
                              hipStream_t stream)
{
    (void)in_sizes; (void)n_in; (void)out_size; (void)ws_size;

    const float* x      = (const float*)d_in[0];
    const float* adj    = (const float*)d_in[1];
    const float* Wg     = (const float*)d_in[2];
    const float* bg     = (const float*)d_in[3];
    const float* pre_w  = (const float*)d_in[4];
    const float* pre_b  = (const float*)d_in[5];
    const float* post_w = (const float*)d_in[6];
    const float* post_b = (const float*)d_in[7];
    const float* qw     = (const float*)d_in[8];
    float* out = (float*)d_out;
    float* Wt  = (float*)d_ws;   // 512*512*4 = 1 MB workspace

    transpose512<<<dim3(16, 16), dim3(32, 8), 0, stream>>>(Wg, Wt);
    gcn_fused_kernel<<<dim3(NN / 16, BB), dim3(256), 0, stream>>>(x, adj, Wt, bg, out);
    quantum_head_kernel<<<dim3(BB), dim3(256), 0, stream>>>(out, pre_w, pre_b, post_w, post_b, qw);
}